// CoreferenceModel_44598940402062
// MI455X (gfx1250) — compile-verified
//
#include <hip/hip_runtime.h>
#include <hip/hip_bf16.h>

typedef __attribute__((ext_vector_type(2))) float v2f;
typedef __attribute__((ext_vector_type(8))) float v8f;

#define D_DIM 768
#define H_DIM 1024
#define K_CAND 250
#define N_VEC 2000
#define M_TOP 400

__device__ __forceinline__ v8f wmma4(v2f a, v2f b, v8f c) {
  // D = A(16x4) * B(4x16) + C(16x16), fp32 WMMA
  return __builtin_amdgcn_wmma_f32_16x16x4_f32(false, a, false, b, (short)0, c,
                                               false, false);
}

__device__ __forceinline__ v8f zero8() {
  v8f z;
#pragma unroll
  for (int q = 0; q < 8; ++q) z[q] = 0.0f;
  return z;
}

// --- CDNA5 async global->LDS copy (ASYNCcnt tracked, bypasses VGPRs) -------
typedef int v4i_vs __attribute__((vector_size(16)));
typedef __attribute__((address_space(1))) v4i_vs g_v4i;
typedef __attribute__((address_space(3))) v4i_vs l_v4i;

__device__ __forceinline__ void async_b128(const float* g, float* l) {
#if __has_builtin(__builtin_amdgcn_global_load_async_to_lds_b128)
  __builtin_amdgcn_global_load_async_to_lds_b128((g_v4i*)g, (l_v4i*)l, 0, 0);
#else
  unsigned lo =
      (unsigned)(uintptr_t)(__attribute__((address_space(3))) char*)(char*)l;
  asm volatile("global_load_async_to_lds_b128 %0, %1, off" ::"v"(lo),
               "v"((const void*)g)
               : "memory");
#endif
}

__device__ __forceinline__ void wait_async0() {
#if __has_builtin(__builtin_amdgcn_s_wait_asynccnt)
  __builtin_amdgcn_s_wait_asynccnt(0);
#else
  asm volatile("s_wait_asynccnt 0x0" ::: "memory");
#endif
}

// ---------------------------------------------------------------------------
// Stage A: scores = relu(V @ Wm1 + bm1) @ wm2 + bm2   (fused, WMMA f32)
// grid: 32 blocks x 256 threads; each block: 64 rows, loops N in chunks of 128
// ---------------------------------------------------------------------------
__global__ __launch_bounds__(256) void mention_score_kernel(
    const float* __restrict__ V, const float* __restrict__ Wm1,
    const float* __restrict__ bm1, const float* __restrict__ wm2,
    const float* __restrict__ bm2, float* __restrict__ scores) {
  __shared__ float As[64 * 20];   // 64 rows x 16 K (stride 20)
  __shared__ float Bs[16 * 144];  // 16 K x 128 N (stride 144)
  __shared__ float rowsum[64];

  const int tid = threadIdx.x;
  const int lane = tid & 31;
  const int wv = tid >> 5;  // wave -> column tile (16 cols)
  const int rowbase = blockIdx.x * 64;
  if (tid < 64) rowsum[tid] = 0.0f;

  const int lr = tid >> 2;        // A-stage row 0..63
  const int lc = (tid & 3) * 4;   // A-stage col 0,4,8,12
  const int br = tid >> 4;        // B-stage row 0..15
  const int bc = (tid & 15) * 8;  // B-stage col 0..120

  const int mrow = lane & 15;
  const int kh = (lane >> 4) << 1;  // K sub-offset {0,2}
  const int hi = (lane >> 4) << 3;  // C/D row offset {0,8}

  int arow = rowbase + lr;
  if (arow > N_VEC - 1) arow = N_VEC - 1;

  float psum[32];
#pragma unroll
  for (int q = 0; q < 32; ++q) psum[q] = 0.0f;

  for (int n0 = 0; n0 < H_DIM; n0 += 128) {
    v8f acc[4];
#pragma unroll
    for (int rt = 0; rt < 4; ++rt) acc[rt] = zero8();

    for (int kc = 0; kc < D_DIM; kc += 16) {
      float4 a4 = *(const float4*)(V + arow * D_DIM + kc + lc);
      *(float4*)(As + lr * 20 + lc) = a4;
      const float* wrow = Wm1 + (kc + br) * H_DIM + n0 + bc;
      *(float4*)(Bs + br * 144 + bc) = *(const float4*)(wrow);
      *(float4*)(Bs + br * 144 + bc + 4) = *(const float4*)(wrow + 4);
      __syncthreads();
#pragma unroll
      for (int ksl = 0; ksl < 4; ++ksl) {
        v2f bf;
        bf.x = Bs[(ksl * 4 + kh) * 144 + wv * 16 + mrow];
        bf.y = Bs[(ksl * 4 + kh + 1) * 144 + wv * 16 + mrow];
#pragma unroll
        for (int rt = 0; rt < 4; ++rt) {
          v2f af;
          af.x = As[(rt * 16 + mrow) * 20 + ksl * 4 + kh];
          af.y = As[(rt * 16 + mrow) * 20 + ksl * 4 + kh + 1];
          acc[rt] = wmma4(af, bf, acc[rt]);
        }
      }
      __syncthreads();
    }
    const int col = n0 + wv * 16 + (lane & 15);
    const float w2 = wm2[col];
    const float bb = bm1[col];
#pragma unroll
    for (int rt = 0; rt < 4; ++rt) {
#pragma unroll
      for (int r = 0; r < 8; ++r) {
        float t = acc[rt][r] + bb;
        t = fmaxf(t, 0.0f);
        psum[rt * 8 + r] += t * w2;
      }
    }
  }
#pragma unroll
  for (int rt = 0; rt < 4; ++rt)
#pragma unroll
    for (int r = 0; r < 8; ++r)
      atomicAdd(&rowsum[rt * 16 + r + hi], psum[rt * 8 + r]);
  __syncthreads();
  if (tid < 64) {
    int row = rowbase + tid;
    if (row < N_VEC) scores[row] = rowsum[tid] + bm2[0];
  }
}

// ---------------------------------------------------------------------------
// Stage B: top-400 by score (stable, desc), then order by (start,end,rank) desc
// ---------------------------------------------------------------------------
__global__ __launch_bounds__(1024) void sort_kernel(
    const float* __restrict__ scores, const int* __restrict__ starts,
    const int* __restrict__ ends, int* __restrict__ perm,
    float* __restrict__ ssel) {
  __shared__ unsigned long long key[2048];
  __shared__ unsigned long long key2[512];
  const int tid = threadIdx.x;
  for (int e = tid; e < 2048; e += 1024) {
    if (e < N_VEC) {
      unsigned ub = __float_as_uint(scores[e]);
      unsigned mapped = (ub & 0x80000000u) ? ~ub : (ub | 0x80000000u);
      key[e] = ((unsigned long long)(~mapped) << 32) | (unsigned)e;
    } else {
      key[e] = ~0ULL;
    }
  }
  __syncthreads();
  for (int k = 2; k <= 2048; k <<= 1) {
    for (int j = k >> 1; j > 0; j >>= 1) {
      for (int e = tid; e < 2048; e += 1024) {
        int l = e ^ j;
        if (l > e) {
          unsigned long long a = key[e], b = key[l];
          bool asc = (e & k) == 0;
          if ((a > b) == asc) {
            key[e] = b;
            key[l] = a;
          }
        }
      }
      __syncthreads();
    }
  }
  if (tid < 512) {
    if (tid < M_TOP) {
      unsigned idx = (unsigned)(key[tid] & 0xFFFFFFFFu);
      unsigned long long st = (unsigned)starts[idx];
      unsigned long long en = (unsigned)ends[idx];
      unsigned long long kk = (st << 40) | (en << 24) |
                              ((unsigned long long)(tid + 1) << 12) | idx;
      key2[tid] = ~kk;
    } else {
      key2[tid] = ~0ULL;
    }
  }
  __syncthreads();
  for (int k = 2; k <= 512; k <<= 1) {
    for (int j = k >> 1; j > 0; j >>= 1) {
      if (tid < 512) {
        int e = tid;
        int l = e ^ j;
        if (l > e) {
          unsigned long long a = key2[e], b = key2[l];
          bool asc = (e & k) == 0;
          if ((a > b) == asc) {
            key2[e] = b;
            key2[l] = a;
          }
        }
      }
      __syncthreads();
    }
  }
  if (tid < M_TOP) {
    unsigned long long kk = ~key2[tid];
    unsigned idx = (unsigned)(kk & 0xFFFu);
    perm[tid] = (int)idx;
    ssel[tid] = scores[idx];
  }
}

__global__ __launch_bounds__(256) void gather_kernel(
    const float* __restrict__ V, const int* __restrict__ perm,
    float* __restrict__ vsel) {
  const int r = blockIdx.x;
  const int src = perm[r];
  for (int c = threadIdx.x; c < D_DIM; c += 256)
    vsel[r * D_DIM + c] = V[src * D_DIM + c];
}

// ---------------------------------------------------------------------------
// Stage C: generic WMMA f32 GEMM  out[M,1024] = X[M,768] @ W[768,1024] (+bias)
// ---------------------------------------------------------------------------
__global__ __launch_bounds__(256) void gemm_kernel(
    const float* __restrict__ X, const float* __restrict__ W,
    const float* __restrict__ bias, float* __restrict__ out, int M) {
  __shared__ float As[32 * 36];
  __shared__ float Bs[32 * 80];
  const int tid = threadIdx.x;
  const int lane = tid & 31;
  const int wv = tid >> 5;
  const int rt = wv >> 2, ct = wv & 3;
  const int nbase = blockIdx.x * 64;
  const int rowbase = blockIdx.y * 32;
  const int lr = tid >> 3;
  const int lc = (tid & 7) * 4;
  const int mrow = lane & 15;
  const int kh = (lane >> 4) << 1;
  const int hi = (lane >> 4) << 3;
  int arow = rowbase + lr;
  if (arow > M - 1) arow = M - 1;

  v8f acc = zero8();
  for (int kc = 0; kc < D_DIM; kc += 32) {
    *(float4*)(As + lr * 36 + lc) = *(const float4*)(X + arow * D_DIM + kc + lc);
    const float* wrow = W + (kc + lr) * H_DIM + nbase + (tid & 7) * 8;
    *(float4*)(Bs + lr * 80 + (tid & 7) * 8) = *(const float4*)(wrow);
    *(float4*)(Bs + lr * 80 + (tid & 7) * 8 + 4) = *(const float4*)(wrow + 4);
    __syncthreads();
#pragma unroll
    for (int ksl = 0; ksl < 8; ++ksl) {
      v2f bf;
      bf.x = Bs[(ksl * 4 + kh) * 80 + ct * 16 + mrow];
      bf.y = Bs[(ksl * 4 + kh + 1) * 80 + ct * 16 + mrow];
      v2f af;
      af.x = As[(rt * 16 + mrow) * 36 + ksl * 4 + kh];
      af.y = As[(rt * 16 + mrow) * 36 + ksl * 4 + kh + 1];
      acc = wmma4(af, bf, acc);
    }
    __syncthreads();
  }
  const int col = nbase + ct * 16 + (lane & 15);
  const float b = bias ? bias[col] : 0.0f;
#pragma unroll
  for (int r = 0; r < 8; ++r) {
    int row = rowbase + rt * 16 + r + hi;
    if (row < M) out[row * H_DIM + col] = acc[r] + b;
  }
}

// ---------------------------------------------------------------------------
// Stage D: ant[i][k] = relu((a_i*v_j)@Wab + aW_i + u_j) . wp2 + bp2 + s_i + s_j
// grid: (399 anchors, 4 k-chunks of 64); 256 threads; the 157-GFLOP kernel.
// Async-to-LDS double-buffered B staging; 2 column tiles per wave (N-chunk 256)
// ---------------------------------------------------------------------------
#define ABUF (64 * 20)
#define BSTR 272
#define BBUF (16 * BSTR)
#define KCHUNKS (D_DIM / 16)

__global__ __launch_bounds__(256) void pair_kernel(
    const float* __restrict__ vsel, const float* __restrict__ Wab,
    const float* __restrict__ aW, const float* __restrict__ u,
    const float* __restrict__ wp2, const float* __restrict__ bp2,
    const float* __restrict__ ssel, float* __restrict__ ant) {
  const int i = blockIdx.x;        // anchor 0..398
  const int k0 = blockIdx.y * 64;  // candidate chunk base
  const int tid = threadIdx.x;
  if (i + 1 + k0 >= M_TOP) {  // entire chunk invalid
    if (tid < 64) {
      int kg = k0 + tid;
      if (kg < K_CAND) ant[i * K_CAND + kg] = -__builtin_inff();
    }
    return;
  }
  __shared__ float Ab[2 * ABUF];  // 64 rows x 16 K, stride 20, double buffered
  __shared__ float Bb[2 * BBUF];  // 16 K x 256 N, stride 272, double buffered
  __shared__ float rowsum[64];
  const int lane = tid & 31;
  const int wv = tid >> 5;
  if (tid < 64) rowsum[tid] = 0.0f;

  const int lr = tid >> 2;         // A-stage row 0..63
  const int lc = (tid & 3) * 4;    // A-stage col 0,4,8,12
  const int br = tid >> 4;         // B-stage row 0..15
  const int bc = (tid & 15) * 16;  // B-stage col 0..240
  const int mrow = lane & 15;
  const int kh = (lane >> 4) << 1;
  const int hi = (lane >> 4) << 3;

  int jrow = i + 1 + k0 + lr;  // candidate rows contiguous after sorting
  if (jrow > M_TOP - 1) jrow = M_TOP - 1;
  const float* arow_p = vsel + i * D_DIM + lc;
  const float* brow_p = vsel + jrow * D_DIM + lc;
  float* Ast = Ab + lr * 20 + lc;

  float psum[32];
#pragma unroll
  for (int q = 0; q < 32; ++q) psum[q] = 0.0f;

  for (int n0 = 0; n0 < H_DIM; n0 += 256) {
    v8f acc[8];  // [ctl*4 + rt]
#pragma unroll
    for (int q = 0; q < 8; ++q) acc[q] = zero8();

    const float* Bsrc = Wab + br * H_DIM + n0 + bc;  // += kc*H_DIM per chunk
    float* Bst = Bb + br * BSTR + bc;
    {  // prologue: stage chunk 0 into buffer 0
#pragma unroll
      for (int q = 0; q < 4; ++q) async_b128(Bsrc + q * 4, Bst + q * 4);
      float4 a4 = *(const float4*)(arow_p);
      float4 b4 = *(const float4*)(brow_p);
      float4 p4;
      p4.x = a4.x * b4.x;
      p4.y = a4.y * b4.y;
      p4.z = a4.z * b4.z;
      p4.w = a4.w * b4.w;
      *(float4*)(Ast) = p4;
      wait_async0();
    }
    __syncthreads();

    for (int c = 0; c < KCHUNKS; ++c) {
      const int cur = c & 1;
      const int nxt = cur ^ 1;
      const bool pf = (c + 1) < KCHUNKS;
      float4 a4n, b4n;
      if (pf) {  // issue async B(c+1) + prefetch A(c+1) before compute(c)
        const int kc = (c + 1) * 16;
        const float* g = Bsrc + kc * H_DIM;
        float* l = Bst + nxt * BBUF;
#pragma unroll
        for (int q = 0; q < 4; ++q) async_b128(g + q * 4, l + q * 4);
        a4n = *(const float4*)(arow_p + kc);
        b4n = *(const float4*)(brow_p + kc);
      }
      const float* Ac = Ab + cur * ABUF;
      const float* Bc = Bb + cur * BBUF;
#pragma unroll
      for (int ksl = 0; ksl < 4; ++ksl) {
        v2f bf0, bf1;
        bf0.x = Bc[(ksl * 4 + kh) * BSTR + (2 * wv) * 16 + mrow];
        bf0.y = Bc[(ksl * 4 + kh + 1) * BSTR + (2 * wv) * 16 + mrow];
        bf1.x = Bc[(ksl * 4 + kh) * BSTR + (2 * wv + 1) * 16 + mrow];
        bf1.y = Bc[(ksl * 4 + kh + 1) * BSTR + (2 * wv + 1) * 16 + mrow];
#pragma unroll
        for (int rt = 0; rt < 4; ++rt) {
          v2f af;
          af.x = Ac[(rt * 16 + mrow) * 20 + ksl * 4 + kh];
          af.y = Ac[(rt * 16 + mrow) * 20 + ksl * 4 + kh + 1];
          acc[rt] = wmma4(af, bf0, acc[rt]);
          acc[4 + rt] = wmma4(af, bf1, acc[4 + rt]);
        }
      }
      if (pf) {  // store prefetched A product into next buffer
        float4 p4;
        p4.x = a4n.x * b4n.x;
        p4.y = a4n.y * b4n.y;
        p4.z = a4n.z * b4n.z;
        p4.w = a4n.w * b4n.w;
        *(float4*)(Ab + nxt * ABUF + lr * 20 + lc) = p4;
      }
      wait_async0();
      __syncthreads();
    }

    // fused epilogue for this 256-col n-chunk
#pragma unroll
    for (int ctl = 0; ctl < 2; ++ctl) {
      const int col = n0 + (2 * wv + ctl) * 16 + (lane & 15);
      const float w2 = wp2[col];
      const float aWv = aW[i * H_DIM + col];  // includes bp1
#pragma unroll
      for (int rt = 0; rt < 4; ++rt) {
#pragma unroll
        for (int r = 0; r < 8; ++r) {
          int lrow = rt * 16 + r + hi;
          int idx = i + 1 + k0 + lrow;
          int j = idx > M_TOP - 1 ? M_TOP - 1 : idx;
          float t = acc[ctl * 4 + rt][r] + aWv + u[j * H_DIM + col];
          t = fmaxf(t, 0.0f);
          psum[rt * 8 + r] += t * w2;
        }
      }
    }
  }
#pragma unroll
  for (int rt = 0; rt < 4; ++rt)
#pragma unroll
    for (int r = 0; r < 8; ++r)
      atomicAdd(&rowsum[rt * 16 + r + hi], psum[rt * 8 + r]);
  __syncthreads();
  if (tid < 64) {
    int kg = k0 + tid;
    if (kg < K_CAND) {
      int idx = i + 1 + kg;
      bool valid = idx < M_TOP;
      int j = valid ? idx : M_TOP - 1;
      float a = rowsum[tid] + bp2[0] + ssel[i] + ssel[j];
      ant[i * K_CAND + kg] = valid ? a : -__builtin_inff();
    }
  }
}

// ---------------------------------------------------------------------------
// Stage E: softmax over 250 logits + implicit trailing 0 logit
// ---------------------------------------------------------------------------
__global__ __launch_bounds__(256) void softmax_kernel(
    const float* __restrict__ ant, float* __restrict__ out) {
  __shared__ float red[256];
  const int r = blockIdx.x;
  const int tid = threadIdx.x;
  float x = (tid < K_CAND) ? ant[r * K_CAND + tid] : -__builtin_inff();
  red[tid] = x;
  __syncthreads();
#pragma unroll
  for (int s = 128; s > 0; s >>= 1) {
    if (tid < s) red[tid] = fmaxf(red[tid], red[tid + s]);
    __syncthreads();
  }
  const float m = fmaxf(red[0], 0.0f);
  __syncthreads();
  float e = (tid < K_CAND) ? expf(x - m) : 0.0f;
  red[tid] = e;
  __syncthreads();
#pragma unroll
  for (int s = 128; s > 0; s >>= 1) {
    if (tid < s) red[tid] += red[tid + s];
    __syncthreads();
  }
  const float ez = expf(0.0f - m);
  const float z = red[0] + ez;
  if (tid < K_CAND) out[r * (K_CAND + 1) + tid] = e / z;
  if (tid == 0) out[r * (K_CAND + 1) + K_CAND] = ez / z;
}

// ---------------------------------------------------------------------------
extern "C" void kernel_launch(void* const* d_in, const int* in_sizes, int n_in,
                              void* d_out, int out_size, void* d_ws,
                              size_t ws_size, hipStream_t stream) {
  const float* vectors = (const float*)d_in[0];
  const float* Wm1 = (const float*)d_in[1];
  const float* bm1 = (const float*)d_in[2];
  const float* wm2 = (const float*)d_in[3];
  const float* bm2 = (const float*)d_in[4];
  const float* Wp1 = (const float*)d_in[5];
  const float* bp1 = (const float*)d_in[6];
  const float* wp2 = (const float*)d_in[7];
  const float* bp2 = (const float*)d_in[8];
  const int* starts = (const int*)d_in[9];
  const int* ends = (const int*)d_in[10];
  // d_in[11] is t (=1000); m = int(0.4*t) = 400, baked into M_TOP.

  char* ws = (char*)d_ws;
  float* scores = (float*)(ws);        // 2000 f
  int* perm = (int*)(ws + 8192);       // 400 i
  float* ssel = (float*)(ws + 10240);  // 400 f
  float* vsel = (float*)(ws + 12288);  // 400x768 f
  float* aW = vsel + M_TOP * D_DIM;    // 400x1024 f (vsel@Wa+bp1)
  float* u = aW + M_TOP * H_DIM;       // 400x1024 f (vsel@Wb)
  float* ant = u + M_TOP * H_DIM;      // 399x250 f

  mention_score_kernel<<<32, 256, 0, stream>>>(vectors, Wm1, bm1, wm2, bm2,
                                               scores);
  sort_kernel<<<1, 1024, 0, stream>>>(scores, starts, ends, perm, ssel);
  gather_kernel<<<M_TOP, 256, 0, stream>>>(vectors, perm, vsel);
  gemm_kernel<<<dim3(16, 13), 256, 0, stream>>>(vsel, Wp1, bp1, aW, M_TOP);
  gemm_kernel<<<dim3(16, 13), 256, 0, stream>>>(vsel, Wp1 + D_DIM * H_DIM,
                                                nullptr, u, M_TOP);
  pair_kernel<<<dim3(M_TOP - 1, 4), 256, 0, stream>>>(
      vsel, Wp1 + 2 * D_DIM * H_DIM, aW, u, wp2, bp2, ssel, ant);
  softmax_kernel<<<M_TOP - 1, 256, 0, stream>>>(ant, (float*)d_out);
}